// DifferentiableAlways_74182675136672
// MI455X (gfx1250) — compile-verified
//
#include <hip/hip_runtime.h>
#include <hip/hip_bf16.h>
#include <stdint.h>

#define T_DIM 4096
#define WAVES_PER_BLOCK 8
#define BLOCK_THREADS (WAVES_PER_BLOCK * 32)
#define LARGE_NUMBER 1.0e6f
#define DELTA 1.0e-3f
#define SCALE 1.0f

typedef unsigned int u32x4 __attribute__((ext_vector_type(4)));
typedef int          i32x8 __attribute__((ext_vector_type(8)));
typedef int          i32x4 __attribute__((ext_vector_type(4)));

__global__ __launch_bounds__(BLOCK_THREADS)
void DifferentiableAlways_kernel(const float* __restrict__ sig,
                                 const int* __restrict__ ts_p,
                                 const int* __restrict__ te_p,
                                 float* __restrict__ out) {
    __shared__ float lds_sig[T_DIM];   // staged by TDM
    __shared__ float lds_m[T_DIM];     // smooth mask, computed on VALU

    const int tid  = threadIdx.x;
    const int lane = tid & 31;
    const int wave = tid >> 5;

    const float t_start = (float)ts_p[0];
    const float t_end   = (float)te_p[0];

    // ---- precompute smooth_time_mask into LDS (overlaps with the TDM DMA below)
    // sigmoid(a)*sigmoid(b) = 1 / ((1+e^-a)(1+e^-b))  -> one v_rcp_f32, no div chains
    #pragma unroll 4
    for (int d = tid; d < T_DIM; d += BLOCK_THREADS) {
        float fd = (float)d;
        float ea = __expf(-SCALE * (fd - t_start));
        float eb = __expf(-SCALE * (t_end - fd));
        lds_m[d] = __builtin_amdgcn_rcpf((1.0f + ea) * (1.0f + eb));
    }

    // ---- TDM: DMA the whole signal (16 KB) global -> LDS, issued by wave 0 ----
    if (wave == 0) {
        uint64_t gaddr = (uint64_t)(uintptr_t)sig;
        uint32_t laddr = (uint32_t)(uintptr_t)(&lds_sig[0]);  // low 32 bits = LDS byte offset

        u32x4 g0;
        g0[0] = 1u;                                   // count=1 (valid), user descriptor
        g0[1] = laddr;                                // lds_addr [63:32]
        g0[2] = (uint32_t)(gaddr & 0xFFFFFFFFu);      // global_addr [95:64]
        g0[3] = (uint32_t)((gaddr >> 32) & 0x01FFFFFFu) | (2u << 30);  // addr hi + type=2

        i32x8 g1;
        g1[0] = 0x00020000;            // workgroup_mask=0, data_size=2 (4 bytes)
        g1[1] = (int)(T_DIM << 16);    // tensor_dim0 = 4096 (bits [63:48])
        g1[2] = 0x00010000;            // tensor_dim1 = 1    (bits [95:80])
        g1[3] = (int)(T_DIM << 16);    // tile_dim0   = 4096 (bits [127:112])
        g1[4] = 1;                     // tile_dim1   = 1
        g1[5] = T_DIM;                 // tensor_dim0_stride = 4096
        g1[6] = 0;
        g1[7] = 0;

        i32x4 gz4 = {0, 0, 0, 0};                     // groups 2/3 unused (<=2D tensor)
        i32x8 gz8 = {0, 0, 0, 0, 0, 0, 0, 0};         // 5th group (clang-23 6-arg form)

        __builtin_amdgcn_tensor_load_to_lds(g0, g1, gz4, gz4, gz8, 0);
        __builtin_amdgcn_s_wait_tensorcnt(0);
    }
    __syncthreads();

    // ---- one column per wave; 32 lanes partition the d-loop ----
    const int c = blockIdx.x * WAVES_PER_BLOCK + wave;   // grid = T_DIM/8 blocks

    // Pass 1: branchless running max of the effective per-tap value
    //   v(d) = (m[d] > delta) ? -x*m : -LARGE   (exactly the reference's masked matrix, negated)
    float mx = -LARGE_NUMBER;
    #pragma unroll 4
    for (int d = lane; d < T_DIM; d += 32) {
        float m   = lds_m[d];
        int   idx = c + d;
        float x   = lds_sig[idx < T_DIM ? idx : (T_DIM - 1)];
        float v   = (m > DELTA) ? (-x * m) : -LARGE_NUMBER;
        mx = fmaxf(mx, v);
    }
    #pragma unroll
    for (int o = 16; o > 0; o >>= 1) mx = fmaxf(mx, __shfl_xor(mx, o, 32));

    // Pass 2: sum exp(v - mx) over all T in-band taps (masked taps underflow to 0,
    // term-for-term identical to the reference logsumexp).
    float s = 0.0f;
    #pragma unroll 4
    for (int d = lane; d < T_DIM; d += 32) {
        float m   = lds_m[d];
        int   idx = c + d;
        float x   = lds_sig[idx < T_DIM ? idx : (T_DIM - 1)];
        float v   = (m > DELTA) ? (-x * m) : -LARGE_NUMBER;
        s += __expf(v - mx);
    }
    #pragma unroll
    for (int o = 16; o > 0; o >>= 1) s += __shfl_xor(s, o, 32);

    if (lane == 0) {
        // T_DIM out-of-band rows of the reference's [2T,T] matrix all sit at LARGE_NUMBER.
        s += (float)T_DIM * __expf(-LARGE_NUMBER - mx);
        out[c] = -(mx + __logf(s));
    }
}

extern "C" void kernel_launch(void* const* d_in, const int* in_sizes, int n_in,
                              void* d_out, int out_size, void* d_ws, size_t ws_size,
                              hipStream_t stream) {
    const float* sig = (const float*)d_in[0];
    const int*   ts  = (const int*)d_in[1];
    const int*   te  = (const int*)d_in[2];
    float*       out = (float*)d_out;

    dim3 grid(T_DIM / WAVES_PER_BLOCK);   // 512 blocks x 8 waves = 4096 waves (1 per column)
    dim3 block(BLOCK_THREADS);
    DifferentiableAlways_kernel<<<grid, block, 0, stream>>>(sig, ts, te, out);
}